// CoordinateVQ_87892210745725
// MI455X (gfx1250) — compile-verified
//
#include <hip/hip_runtime.h>

typedef __attribute__((ext_vector_type(2))) float v2f;
typedef __attribute__((ext_vector_type(8))) float v8f;

#define COORD_SCALE_XY 1e-5f
#define COORD_SCALE_Z  0.3f
#define COMMITMENT_COST 0.25f

#define K_CODES 512
#define GROUPS  32           // 512 / 16 codewords per WMMA tile
#define TOKENS  131072       // 16 * 8192
#define THREADS 256          // 8 waves (wave32)
#define WAVES_PER_BLOCK 8
#define TOKENS_PER_WAVE 16
#define BLOCKS  1024         // 131072 / (8*16)

// One wave handles a 16-token tile. Distances to all 512 codewords are
// computed as 32x V_WMMA_F32_16X16X4_F32 with A = [sx,sy,sz,1] rows and
// B = [-2cx,-2cy,-2cz,|c|^2] columns, so D[m][n] = |c_n|^2 - 2 x_m.c_n,
// which has the same argmin as the squared L2 distance.
__global__ __launch_bounds__(THREADS)
void CoordinateVQ_main_kernel(const float* __restrict__ coords,
                              const int*   __restrict__ mask,
                              const float* __restrict__ cb,
                              float*       __restrict__ out,
                              float*       __restrict__ partials) {
    // B-tiles in WMMA per-lane layout: [group][vgpr][lane]
    //   vgpr0: lanes 0-15 = K=0 (-2cx), lanes 16-31 = K=2 (-2cz)
    //   vgpr1: lanes 0-15 = K=1 (-2cy), lanes 16-31 = K=3 (|c|^2)
    __shared__ float bTile[GROUPS][2][32];   // 8 KB
    __shared__ float cbS[K_CODES * 3];       // 6 KB raw codebook for gather
    __shared__ float waveSe[WAVES_PER_BLOCK];
    __shared__ float waveCnt[WAVES_PER_BLOCK];

    const int tid  = threadIdx.x;
    const int lane = tid & 31;
    const int wv   = tid >> 5;

    // Stage codebook into LDS (2 codewords per thread).
    for (int k = tid; k < K_CODES; k += THREADS) {
        float cx = cb[k * 3 + 0], cy = cb[k * 3 + 1], cz = cb[k * 3 + 2];
        cbS[k * 3 + 0] = cx; cbS[k * 3 + 1] = cy; cbS[k * 3 + 2] = cz;
        int g = k >> 4, n = k & 15;
        bTile[g][0][n]      = -2.0f * cx;            // K=0
        bTile[g][1][n]      = -2.0f * cy;            // K=1
        bTile[g][0][16 + n] = -2.0f * cz;            // K=2
        bTile[g][1][16 + n] = cx*cx + cy*cy + cz*cz; // K=3
    }
    __syncthreads();

    const int gw  = blockIdx.x * WAVES_PER_BLOCK + wv;
    const int t0  = gw * TOKENS_PER_WAVE;
    const int m   = lane & 15;
    const int tok = t0 + m;

    // Each lane loads the coords of token t0 + (lane&15); lanes 0-15 feed
    // K=0/1 of the A matrix, lanes 16-31 feed K=2/3 of the same rows.
    float rx = coords[tok * 3 + 0];
    float ry = coords[tok * 3 + 1];
    float rz = coords[tok * 3 + 2];
    float sx = rx * COORD_SCALE_XY;
    float sy = ry * COORD_SCALE_XY;
    float sz = rz * COORD_SCALE_Z;

    v2f a;
    if (lane < 16) { a.x = sx; a.y = sy; }
    else           { a.x = sz; a.y = 1.0f; }

    v8f best;
    int bidx[8];
#pragma unroll
    for (int r = 0; r < 8; ++r) { best[r] = 3.4e38f; bidx[r] = 0; }

    // Scan all 512 codewords: one WMMA per 16-codeword group.
    // Strict '<' suffices here: candidates arrive in increasing codeword
    // order, so on ties the incumbent (lower index) wins — same as argmin.
    for (int g = 0; g < GROUPS; ++g) {
        v2f b;
        b.x = bTile[g][0][lane];
        b.y = bTile[g][1][lane];
        v8f c = {};
        c = __builtin_amdgcn_wmma_f32_16x16x4_f32(
                /*neg_a=*/false, a, /*neg_b=*/false, b,
                /*c_mod=*/(short)0, c, /*reuse_a=*/false, /*reuse_b=*/false);
        const int cw = (g << 4) | m;  // codeword index this lane's column holds
#pragma unroll
        for (int r = 0; r < 8; ++r) {
            bool lt = c[r] < best[r];
            best[r] = lt ? c[r] : best[r];
            bidx[r] = lt ? cw   : bidx[r];
        }
    }

    // Butterfly min-reduce across the 16 lanes sharing each (row, half).
    // xor masks < 16 keep lanes within their half (0-15 / 16-31).
    // Indices across lanes are unordered, so tie-break on lower index here.
#pragma unroll
    for (int off = 1; off < 16; off <<= 1) {
#pragma unroll
        for (int r = 0; r < 8; ++r) {
            float os = __shfl_xor(best[r], off, 32);
            int   oi = __shfl_xor(bidx[r], off, 32);
            bool take = (os < best[r]) || ((os == best[r]) & (oi < bidx[r]));
            best[r] = take ? os : best[r];
            bidx[r] = take ? oi : bidx[r];
        }
    }

    // C layout: VGPR r, lanes 0-15 -> token M=r; lanes 16-31 -> token M=r+8.
    // Writer lanes chosen so each writer's own loaded coords match its token:
    // lane r writes token t0+r; lane 24+r (m=r+8) writes token t0+r+8.
    float seLocal = 0.0f, cntLocal = 0.0f;
#pragma unroll
    for (int r = 0; r < 8; ++r) {
        if (lane == r || lane == 24 + r) {
            int  idx   = bidx[r];
            bool has   = (fabsf(rx) + fabsf(ry) + fabsf(rz)) > 0.0f;
            bool valid = has && (mask[tok] != 0);
            float qx = cbS[idx * 3 + 0];
            float qy = cbS[idx * 3 + 1];
            float qz = cbS[idx * 3 + 2];
            float ox, oy, oz;
            if (valid) { ox = qx / COORD_SCALE_XY; oy = qy / COORD_SCALE_XY; oz = qz / COORD_SCALE_Z; }
            else       { ox = sx / COORD_SCALE_XY; oy = sy / COORD_SCALE_XY; oz = sz / COORD_SCALE_Z; }
            out[tok * 3 + 0] = ox;
            out[tok * 3 + 1] = oy;
            out[tok * 3 + 2] = oz;
            out[TOKENS * 3 + 1 + tok] = valid ? (float)idx : 0.0f;
            if (valid) {
                float dx = sx - qx, dy = sy - qy, dz = sz - qz;
                seLocal  = dx * dx + dy * dy + dz * dz;
                cntLocal = 1.0f;
            }
        }
    }

    // Deterministic block partial: wave butterfly sum -> LDS -> thread-0 sum.
#pragma unroll
    for (int off = 16; off >= 1; off >>= 1) {
        seLocal  += __shfl_xor(seLocal,  off, 32);
        cntLocal += __shfl_xor(cntLocal, off, 32);
    }
    if (lane == 0) { waveSe[wv] = seLocal; waveCnt[wv] = cntLocal; }
    __syncthreads();
    if (tid == 0) {
        float se = 0.0f, cnt = 0.0f;
        for (int w = 0; w < WAVES_PER_BLOCK; ++w) { se += waveSe[w]; cnt += waveCnt[w]; }
        partials[blockIdx.x * 2 + 0] = se;
        partials[blockIdx.x * 2 + 1] = cnt;
    }
}

// Fixed-order final reduction -> scalar loss (bit-deterministic per launch).
__global__ void CoordinateVQ_loss_kernel(const float* __restrict__ partials,
                                         float* __restrict__ out) {
    if (blockIdx.x == 0 && threadIdx.x == 0) {
        float se = 0.0f, cnt = 0.0f;
        for (int i = 0; i < BLOCKS; ++i) {
            se  += partials[2 * i + 0];
            cnt += partials[2 * i + 1];
        }
        if (cnt < 1.0f) cnt = 1.0f;
        out[TOKENS * 3] = COMMITMENT_COST * se / (cnt * 3.0f);
    }
}

extern "C" void kernel_launch(void* const* d_in, const int* in_sizes, int n_in,
                              void* d_out, int out_size, void* d_ws, size_t ws_size,
                              hipStream_t stream) {
    const float* coords = (const float*)d_in[0];   // [16,8192,3] f32
    const int*   mask   = (const int*)d_in[1];     // [16,8192]   i32
    const float* cb     = (const float*)d_in[2];   // [512,3]     f32
    float* out      = (float*)d_out;               // [393216 q | 1 loss | 131072 idx]
    float* partials = (float*)d_ws;                // 1024 * 2 floats

    CoordinateVQ_main_kernel<<<BLOCKS, THREADS, 0, stream>>>(coords, mask, cb, out, partials);
    CoordinateVQ_loss_kernel<<<1, 32, 0, stream>>>(partials, out);
}